// APPNP_layer_14491219657227
// MI455X (gfx1250) — compile-verified
//
#include <hip/hip_runtime.h>
#include <stdint.h>

#define D_FEAT          128
#define K_STEPS         10
#define ALPHA_F         0.1f
#define KEEP_SCALE_F    2.0f      // 1/(1-0.5)
#define ROWS_PER_BLOCK  8
#define MAX_EB          1024      // staged edges per block (avg is ~128)

// ---------------------------------------------------------------- utilities

__device__ __forceinline__ unsigned mix32(unsigned h) {
  h ^= h >> 16; h *= 0x7feb352dU;
  h ^= h >> 15; h *= 0x846ca68bU;
  h ^= h >> 16; return h;
}

// CDNA5 async copy global -> LDS (tracked by ASYNCcnt).
// LDS aperture flat addresses carry the LDS byte offset in the low 32 bits
// (ISA 10.2), so truncating the generic pointer is the
// wave-relative LDS address the instruction wants.
__device__ __forceinline__ void async_ld_b32(const void* gsrc, void* lds_dst) {
  asm volatile("global_load_async_to_lds_b32 %0, %1, off"
               :: "v"((unsigned)(uintptr_t)lds_dst), "v"(gsrc)
               : "memory");
}

__device__ __forceinline__ void wait_async_zero() {
#if __has_builtin(__builtin_amdgcn_s_wait_asynccnt)
  __builtin_amdgcn_s_wait_asynccnt(0);
#else
  asm volatile("s_wait_asynccnt 0" ::: "memory");
#endif
}

__device__ __forceinline__ float readfirstlane_f32(float v) {
  return __uint_as_float(__builtin_amdgcn_readfirstlane(__float_as_uint(v)));
}

// ---------------------------------------------------------------- CSR build

__global__ void zero_counts_kernel(int* __restrict__ rowptr,
                                   int* __restrict__ fill, int n_nodes) {
  int i = blockIdx.x * blockDim.x + threadIdx.x;
  if (i <= n_nodes) rowptr[i] = 0;
  if (i <  n_nodes) fill[i]   = 0;
}

__global__ void edge_hist_kernel(const int* __restrict__ rows,
                                 int* __restrict__ rowptr, int n_edges) {
  int e = blockIdx.x * blockDim.x + threadIdx.x;
  if (e < n_edges) atomicAdd(&rowptr[rows[e] + 1], 1);
}

// Single-block inclusive scan over rowptr[0..n-1] (counts stored at +1).
// wave32 shfl-scan per wave + wave-0 scan of the 32 wave sums:
// 4 block barriers per 1024-chunk instead of ~20.
__global__ __launch_bounds__(1024)
void scan_kernel(int* __restrict__ rowptr, int n) {
  __shared__ int wsum[32];
  __shared__ int carry_s;
  const int tid  = threadIdx.x;
  const int lane = tid & 31;
  const int wid  = tid >> 5;
  if (tid == 0) carry_s = 0;
  __syncthreads();

  for (int base = 0; base < n; base += 1024) {
    const int i = base + tid;
    int v = (i < n) ? rowptr[i] : 0;

    // inclusive scan within the wave (wave32)
    int s = v;
    #pragma unroll
    for (int off = 1; off < 32; off <<= 1) {
      int t = __shfl_up(s, off, 32);
      if (lane >= off) s += t;
    }
    if (lane == 31) wsum[wid] = s;
    __syncthreads();

    // wave 0 scans the 32 wave totals
    if (wid == 0) {
      int ws = wsum[lane];
      #pragma unroll
      for (int off = 1; off < 32; off <<= 1) {
        int t = __shfl_up(ws, off, 32);
        if (lane >= off) ws += t;
      }
      wsum[lane] = ws;  // inclusive wave-prefix
    }
    __syncthreads();

    const int waveBase = (wid > 0) ? wsum[wid - 1] : 0;
    const int out = s + waveBase + carry_s;
    if (i < n) rowptr[i] = out;
    __syncthreads();                 // all reads of carry_s / wsum done
    if (tid == 1023) carry_s = out;  // chunk total (zero-padded OOB lanes)
    __syncthreads();
  }
}

__global__ void edge_scatter_kernel(const int* __restrict__ rows,
                                    const int* __restrict__ cols,
                                    const float* __restrict__ vals,
                                    const int* __restrict__ rowptr,
                                    int* __restrict__ fill,
                                    int* __restrict__ col_s,
                                    float* __restrict__ w_s,
                                    int* __restrict__ eid_s, int n_edges) {
  int e = blockIdx.x * blockDim.x + threadIdx.x;
  if (e >= n_edges) return;
  int r   = rows[e];
  int pos = rowptr[r] + atomicAdd(&fill[r], 1);
  col_s[pos] = cols[e];
  w_s[pos]   = vals[e] * (KEEP_SCALE_F * (1.0f - ALPHA_F));  // fold scales once
  eid_s[pos] = e;
}

// Make per-row ordering deterministic (atomic placement order varies between
// replays; fp accumulation order must not). Avg segment is 16 entries.
__global__ void sort_rows_kernel(const int* __restrict__ rowptr,
                                 int* __restrict__ col_s,
                                 float* __restrict__ w_s,
                                 int* __restrict__ eid_s, int n_nodes) {
  int r = blockIdx.x * blockDim.x + threadIdx.x;
  if (r >= n_nodes) return;
  int b = rowptr[r], e = rowptr[r + 1];
  for (int i = b + 1; i < e; ++i) {
    int ke = eid_s[i]; int kc = col_s[i]; float kw = w_s[i];
    int j = i - 1;
    while (j >= b && eid_s[j] > ke) {
      eid_s[j + 1] = eid_s[j];
      col_s[j + 1] = col_s[j];
      w_s[j + 1]   = w_s[j];
      --j;
    }
    eid_s[j + 1] = ke; col_s[j + 1] = kc; w_s[j + 1] = kw;
  }
}

// ---------------------------------------------------------------- propagate
// Block = 128 threads = 4 wave32s. Each wave owns one destination row at a
// time (lane -> float4 of the 128-wide feature row => global_load_b128,
// 512B coalesced per edge). The block's CSR slice is contiguous, so it is
// staged to LDS with CDNA5 async copies up front.

__global__ __launch_bounds__(128)
void propagate_step_kernel(const float* __restrict__ Hprev,
                           const float* __restrict__ x,
                           float* __restrict__ Hnext,
                           const int* __restrict__ rowptr,
                           const int* __restrict__ col_s,
                           const float* __restrict__ w_s,
                           const int* __restrict__ eid_s,
                           int n_nodes, unsigned step) {
  __shared__ int   s_col[MAX_EB];
  __shared__ float s_w[MAX_EB];
  __shared__ int   s_eid[MAX_EB];

  const int tid  = threadIdx.x;
  const int wave = tid >> 5;
  const int lane = tid & 31;
  const int r0   = blockIdx.x * ROWS_PER_BLOCK;
  int nrows = n_nodes - r0;
  if (nrows > ROWS_PER_BLOCK) nrows = ROWS_PER_BLOCK;

  const int e0  = rowptr[r0];
  const int e1  = rowptr[r0 + nrows];
  const int cnt = e1 - e0;
  const bool inLDS = (cnt <= MAX_EB);

  if (inLDS) {
    for (int i = tid; i < cnt; i += 128) {
      async_ld_b32(col_s + e0 + i, &s_col[i]);
      async_ld_b32(w_s   + e0 + i, &s_w[i]);
      async_ld_b32(eid_s + e0 + i, &s_eid[i]);
    }
    wait_async_zero();
  }
  __syncthreads();

  const unsigned salt = 0x9E3779B9u * (step + 1u);

  for (int rr = wave; rr < nrows; rr += 4) {
    const int r  = r0 + rr;
    const int jb = rowptr[r]     - e0;
    const int je = rowptr[r + 1] - e0;

    float4 acc = make_float4(0.f, 0.f, 0.f, 0.f);
    for (int j = jb; j < je; ++j) {
      int eid = inLDS ? s_eid[j] : eid_s[e0 + j];
      eid = __builtin_amdgcn_readfirstlane(eid);
      const unsigned h = mix32((unsigned)eid ^ salt);   // p(keep)=0.5
      if (h >> 31) {                                    // wave-uniform branch
        int c = inLDS ? s_col[j] : col_s[e0 + j];
        c = __builtin_amdgcn_readfirstlane(c);
        float wv = inLDS ? s_w[j] : w_s[e0 + j];
        wv = readfirstlane_f32(wv);
        // hide L2 gather latency of the next edge's source row
        if (j + 1 < je) {
          int cn = inLDS ? s_col[j + 1] : col_s[e0 + j + 1];
          cn = __builtin_amdgcn_readfirstlane(cn);
          __builtin_prefetch(Hprev + (size_t)cn * D_FEAT + 4 * lane);
        }
        const float4 hv =
            *(const float4*)(Hprev + (size_t)c * D_FEAT + 4 * lane);
        acc.x = fmaf(wv, hv.x, acc.x);
        acc.y = fmaf(wv, hv.y, acc.y);
        acc.z = fmaf(wv, hv.z, acc.z);
        acc.w = fmaf(wv, hv.w, acc.w);
      }
    }

    const float4 xv = *(const float4*)(x + (size_t)r * D_FEAT + 4 * lane);
    float4 o;
    o.x = fmaf(ALPHA_F, xv.x, acc.x);
    o.y = fmaf(ALPHA_F, xv.y, acc.y);
    o.z = fmaf(ALPHA_F, xv.z, acc.z);
    o.w = fmaf(ALPHA_F, xv.w, acc.w);
    *(float4*)(Hnext + (size_t)r * D_FEAT + 4 * lane) = o;
  }
}

// ---------------------------------------------------------------- launcher

extern "C" void kernel_launch(void* const* d_in, const int* in_sizes, int n_in,
                              void* d_out, int out_size, void* d_ws, size_t ws_size,
                              hipStream_t stream) {
  const float* x    = (const float*)d_in[0];
  const int*   rows = (const int*)  d_in[1];
  const int*   cols = (const int*)  d_in[2];
  const float* vals = (const float*)d_in[3];

  const int nN = in_sizes[0] / D_FEAT;
  const int nE = in_sizes[1];

  // Carve workspace (256B-aligned chunks).
  size_t off = 0;
  auto take = [&](size_t bytes) -> size_t {
    size_t o = off;
    off += (bytes + 255) & ~(size_t)255;
    return o;
  };
  char* w = (char*)d_ws;
  float* Ha     = (float*)(w + take((size_t)nN * D_FEAT * sizeof(float)));
  int*   rowptr = (int*)  (w + take((size_t)(nN + 1) * sizeof(int)));
  int*   fill   = (int*)  (w + take((size_t)nN * sizeof(int)));
  int*   col_s  = (int*)  (w + take((size_t)nE * sizeof(int)));
  float* w_s    = (float*)(w + take((size_t)nE * sizeof(float)));
  int*   eid_s  = (int*)  (w + take((size_t)nE * sizeof(int)));
  if (off > ws_size) return;  // cannot run without scratch

  // --- build deterministic CSR (once per launch, captured in the graph) ---
  zero_counts_kernel<<<(nN + 1 + 255) / 256, 256, 0, stream>>>(rowptr, fill, nN);
  edge_hist_kernel<<<(nE + 255) / 256, 256, 0, stream>>>(rows, rowptr, nE);
  scan_kernel<<<1, 1024, 0, stream>>>(rowptr, nN + 1);
  edge_scatter_kernel<<<(nE + 255) / 256, 256, 0, stream>>>(
      rows, cols, vals, rowptr, fill, col_s, w_s, eid_s, nE);
  sort_rows_kernel<<<(nN + 255) / 256, 256, 0, stream>>>(
      rowptr, col_s, w_s, eid_s, nN);

  // --- K propagation steps, ping-pong ws <-> d_out (K even => ends in d_out)
  const float* src = x;
  float* outp = (float*)d_out;
  const int pgrid = (nN + ROWS_PER_BLOCK - 1) / ROWS_PER_BLOCK;
  for (int s = 0; s < K_STEPS; ++s) {
    float* dst = (s & 1) ? outp : Ha;
    propagate_step_kernel<<<pgrid, 128, 0, stream>>>(
        src, x, dst, rowptr, col_s, w_s, eid_s, nN, (unsigned)s);
    src = dst;
  }
}